// RobertaHybridEncoder_81363860456165
// MI455X (gfx1250) — compile-verified
//
#include <hip/hip_runtime.h>
#include <math.h>

#define B_  4
#define S_  512
#define H_  768
#define D_  24
#define M_  96
#define O_  768

typedef float v2f __attribute__((ext_vector_type(2)));
typedef float v8f __attribute__((ext_vector_type(8)));

// ---------------------------------------------------------------------------
// CDNA5 async global->LDS copy (ASYNCcnt-tracked, no VGPR data path).
// VDST = per-lane LDS byte offset, VADDR = 64-bit global address.
// ---------------------------------------------------------------------------
__device__ __forceinline__ void async_g2l_b32(unsigned lds_off, const void* gptr)
{
    asm volatile("global_load_async_to_lds_b32 %0, %1, off"
                 :: "v"(lds_off), "v"((unsigned long long)(uintptr_t)gptr)
                 : "memory");
}
__device__ __forceinline__ void wait_async0()
{
    asm volatile("s_wait_asynccnt 0x0" ::: "memory");
}
__device__ __forceinline__ unsigned lds_off_of(const void* p)
{
    // generic LDS address carries the wave-relative LDS offset in its low 32 bits
    return (unsigned)(uintptr_t)p;
}

// ---------------------------------------------------------------------------
// Kernel 1: Zj = H_j @ Wpj ; Zi = H_i @ Wpi ; tj = Zj @ W1j ; ti = Zi @ W1i
// grid = (B*S, 2), block = 256.  blockIdx.y: 0 -> j side, 1 -> i side
// ---------------------------------------------------------------------------
__global__ void k_proj(const float* __restrict__ Hj, const float* __restrict__ Hi,
                       const float* __restrict__ Wpj, const float* __restrict__ Wpi,
                       const float* __restrict__ Ws1,
                       float* __restrict__ Zj, float* __restrict__ Zi,
                       float* __restrict__ tj, float* __restrict__ ti)
{
    __shared__ float row[H_];
    __shared__ float part[8 * D_];
    __shared__ float zrow[D_];
    const int r    = blockIdx.x;          // 0 .. B*S-1
    const int side = blockIdx.y;          // 0 = j, 1 = i
    const int tid  = threadIdx.x;

    const float* src = side ? Hi : Hj;
    const float* Wp  = side ? Wpi : Wpj;
    const float* W1  = Ws1 + side * D_ * M_;      // rows [0..23] or [24..47]
    float* Zout = side ? Zi : Zj;
    float* tout = side ? ti : tj;

    for (int h = tid; h < H_; h += 256) row[h] = src[(size_t)r * H_ + h];
    __syncthreads();

    // 192 threads: 24 columns x 8 segments of 96
    if (tid < 8 * D_) {
        const int d = tid % D_, seg = tid / D_;
        float acc = 0.f;
        const int h0 = seg * 96;
        #pragma unroll 4
        for (int h = h0; h < h0 + 96; ++h) acc = fmaf(row[h], Wp[h * D_ + d], acc);
        part[seg * D_ + d] = acc;
    }
    __syncthreads();

    if (tid < D_) {
        float acc = 0.f;
        #pragma unroll
        for (int s = 0; s < 8; ++s) acc += part[s * D_ + tid];
        zrow[tid] = acc;
        Zout[(size_t)r * D_ + tid] = acc;
    }
    __syncthreads();

    if (tid < M_) {
        float acc = 0.f;
        #pragma unroll
        for (int d = 0; d < D_; ++d) acc = fmaf(zrow[d], W1[d * M_ + tid], acc);
        tout[(size_t)r * M_ + tid] = acc;
    }
}

// ---------------------------------------------------------------------------
// Kernel 2: pairwise logits + softmax -> probs[b,p,q]   (WMMA fp32 version)
//
// For fixed (b,p):  hid[q,m] = relu( UV[q,:] @ Wcat[:,m] + tj[m] + ti[q,m] + bs1[m] )
//                   logit[q] = hid[q,:] . ws2 + bs2 + mask
// UV[q][k]: k<24 -> Zj[p,d]*Zi[q,d],  k>=24 -> |Zj[p,d]-Zi[q,d]|   (K = 48)
// grid = B*S blocks, block = 256 (8 waves). q tiled by 64; each wave owns
// a 16(q) x 48(m) slab => 3 accumulators, K-loop of 12 WMMA steps.
// ---------------------------------------------------------------------------
#define UVS 50                          // LDS row stride (floats, even => 8B pairs)
__global__ void k_pair(const float* __restrict__ Zj, const float* __restrict__ Zi,
                       const float* __restrict__ tj, const float* __restrict__ ti,
                       const float* __restrict__ Ws1, const float* __restrict__ bs1,
                       const float* __restrict__ ws2, const float* __restrict__ bs2_p,
                       const int* __restrict__ attn_mask,
                       float* __restrict__ probs)
{
    __shared__ __align__(16) float UV[64 * UVS];    // UV[q][k]
    __shared__ __align__(16) float Wt[M_ * UVS];    // Wcat transposed: Wt[m][k]
    __shared__ float tj_s[M_];
    __shared__ float bs1_s[M_];
    __shared__ float ws2_s[M_];
    __shared__ float zj_s[D_];
    __shared__ float logit_s[S_];
    __shared__ float red[256];

    const int tid  = threadIdx.x;
    const int lane = tid & 31;
    const int wid  = tid >> 5;          // 0..7
    const int qsub = wid & 3;           // 16-row q subtile
    const int mh   = wid >> 2;          // 0/1 -> m half (48 cols)
    const int l15  = lane & 15;
    const int koff = (lane < 16) ? 0 : 2;

    const int b = blockIdx.x / S_;
    const int p = blockIdx.x % S_;

    const float* W1h = Ws1 + 2 * D_ * M_;   // rows 48..71
    const float* W1d = Ws1 + 3 * D_ * M_;   // rows 72..95

    // Wcat transposed into LDS: Wt[m][k], k in [0,48)
    for (int i = tid; i < 2 * D_ * M_; i += 256) {
        const int k = i / M_, m = i % M_;
        const float w = (k < D_) ? W1h[k * M_ + m] : W1d[(k - D_) * M_ + m];
        Wt[m * UVS + k] = w;
    }
    if (tid < M_) {
        tj_s[tid]  = tj[((size_t)b * S_ + p) * M_ + tid];
        bs1_s[tid] = bs1[tid];
        ws2_s[tid] = ws2[tid];
    }
    if (tid < D_) zj_s[tid] = Zj[((size_t)b * S_ + p) * D_ + tid];
    logit_s[tid]       = 0.f;
    logit_s[tid + 256] = 0.f;
    __syncthreads();

    for (int q0 = 0; q0 < S_; q0 += 64) {
        // build UV tile (64 x 48)
        for (int idx = tid; idx < 64 * D_; idx += 256) {
            const int q = idx / D_, d = idx % D_;
            const float zi = Zi[((size_t)b * S_ + q0 + q) * D_ + d];
            UV[q * UVS + d]      = zj_s[d] * zi;
            UV[q * UVS + D_ + d] = fabsf(zj_s[d] - zi);
        }
        __syncthreads();

        v8f acc0{}; v8f acc1{}; v8f acc2{};
        #pragma unroll
        for (int kk = 0; kk < 48; kk += 4) {
            v2f a  = *(const v2f*)&UV[(qsub * 16 + l15) * UVS + kk + koff];
            v2f b0 = *(const v2f*)&Wt[(mh * 48 +  0 + l15) * UVS + kk + koff];
            v2f b1 = *(const v2f*)&Wt[(mh * 48 + 16 + l15) * UVS + kk + koff];
            v2f b2 = *(const v2f*)&Wt[(mh * 48 + 32 + l15) * UVS + kk + koff];
            acc0 = __builtin_amdgcn_wmma_f32_16x16x4_f32(false, a, false, b0, (short)0, acc0, false, false);
            acc1 = __builtin_amdgcn_wmma_f32_16x16x4_f32(false, a, false, b1, (short)0, acc1, false, false);
            acc2 = __builtin_amdgcn_wmma_f32_16x16x4_f32(false, a, false, b2, (short)0, acc2, false, false);
        }

        // fused epilogue: relu(+tj+ti+bs1) . ws2, row-reduce over m lanes
        const int m0 = mh * 48 + l15;
        const float t0 = tj_s[m0] + bs1_s[m0];
        const float t1 = tj_s[m0 + 16] + bs1_s[m0 + 16];
        const float t2 = tj_s[m0 + 32] + bs1_s[m0 + 32];
        const float w0 = ws2_s[m0], w1 = ws2_s[m0 + 16], w2 = ws2_s[m0 + 32];
        #pragma unroll
        for (int i = 0; i < 8; ++i) {
            const int q = q0 + qsub * 16 + i + ((lane < 16) ? 0 : 8);
            const float* tirow = ti + ((size_t)b * S_ + q) * M_;
            float h0 = fmaxf(acc0[i] + t0 + tirow[m0],      0.f);
            float h1 = fmaxf(acc1[i] + t1 + tirow[m0 + 16], 0.f);
            float h2 = fmaxf(acc2[i] + t2 + tirow[m0 + 32], 0.f);
            float c  = h0 * w0;
            c = fmaf(h1, w1, c);
            c = fmaf(h2, w2, c);
            // butterfly reduce within each 16-lane half
            c += __shfl_xor(c, 1);
            c += __shfl_xor(c, 2);
            c += __shfl_xor(c, 4);
            c += __shfl_xor(c, 8);
            if (l15 == 0) atomicAdd(&logit_s[q], c);   // two m-halves accumulate
        }
        __syncthreads();   // atomics done before UV is rebuilt
    }

    // softmax over 512 logits (2 per thread)
    const float bs2 = bs2_p[0];
    const int q0 = tid, q1 = tid + 256;
    float lg0 = logit_s[q0] + bs2 + (1.0f - (float)attn_mask[b * S_ + q0]) * (-3.402823466e38f);
    float lg1 = logit_s[q1] + bs2 + (1.0f - (float)attn_mask[b * S_ + q1]) * (-3.402823466e38f);

    red[tid] = fmaxf(lg0, lg1);
    __syncthreads();
    for (int s = 128; s > 0; s >>= 1) {
        if (tid < s) red[tid] = fmaxf(red[tid], red[tid + s]);
        __syncthreads();
    }
    const float gmax = red[0];
    __syncthreads();

    const float e0 = __expf(lg0 - gmax);
    const float e1 = __expf(lg1 - gmax);
    red[tid] = e0 + e1;
    __syncthreads();
    for (int s = 128; s > 0; s >>= 1) {
        if (tid < s) red[tid] += red[tid + s];
        __syncthreads();
    }
    const float inv = 1.0f / red[0];

    float* prow = probs + ((size_t)b * S_ + p) * S_;
    prow[q0] = e0 * inv;
    prow[q1] = e1 * inv;
}

// ---------------------------------------------------------------------------
// WMMA fp32 GEMM core: 64x64 block tile, 8 waves (4 in M x 2 in N),
// each wave 16x32 via V_WMMA_F32_16X16X4_F32.
// Double-buffered LDS; B tiles streamed with GLOBAL_LOAD_ASYNC_TO_LDS
// (per-lane LDS scatter does the transpose), A prefetched into registers
// and sunk to LDS after the WMMA block. One barrier per K-slab.
// ---------------------------------------------------------------------------
struct AFetchPlain {
    const float* A; int lda;
    __device__ __forceinline__ float operator()(int r, int k) const { return A[(size_t)r * lda + k]; }
};
struct AFetchFused {   // [ctx | H_j | ctx*H_j], K = 3*H
    const float* ctx; const float* hj;
    __device__ __forceinline__ float operator()(int r, int k) const {
        if (k < H_)          return ctx[(size_t)r * H_ + k];
        else if (k < 2 * H_) return hj[(size_t)r * H_ + (k - H_)];
        int kk = k - 2 * H_;
        return ctx[(size_t)r * H_ + kk] * hj[(size_t)r * H_ + kk];
    }
};
struct EpiStore {
    float* out; int ldc;
    __device__ __forceinline__ void operator()(int r, int c, float v) const { out[(size_t)r * ldc + c] = v; }
};
struct EpiBiasRelu {
    float* out; const float* bias;
    __device__ __forceinline__ void operator()(int r, int c, float v) const {
        out[(size_t)r * O_ + c] = fmaxf(v + bias[c], 0.f);
    }
};
struct EpiBiasAlpha {
    float* out; const float* bias; const float* alpha;
    __device__ __forceinline__ void operator()(int r, int c, float v) const {
        out[(size_t)r * H_ + c] = alpha[0] * (v + bias[c]);
    }
};

template <typename AF, typename EPI>
__device__ __forceinline__ void gemm_tile_64x64(AF fA, const float* __restrict__ Bg, int ldb,
                                                int Ktot, int row0, int col0, EPI epi)
{
    __shared__ __align__(16) float As[2][64 * 34];   // As[i][k], row stride 34
    __shared__ __align__(16) float Bs[2][64 * 34];   // transposed: Bs[n][k]

    const int tid  = threadIdx.x;
    const int lane = tid & 31;
    const int wid  = tid >> 5;        // 0..7
    const int wm   = wid & 3;         // M subtile (16 rows)
    const int wn   = wid >> 2;        // N subtile (32 cols)
    const int l15  = lane & 15;
    const int koff = (lane < 16) ? 0 : 2;

    v8f acc0{}; v8f acc1{};

    // ---- prologue: stage slab 0 ----
    for (int idx = tid; idx < 64 * 32; idx += 256) {
        int i = idx >> 5, k = idx & 31;
        As[0][i * 34 + k] = fA(row0 + i, k);
    }
    for (int idx = tid; idx < 64 * 32; idx += 256) {
        int n = idx & 63, k = idx >> 6;
        async_g2l_b32(lds_off_of(&Bs[0][n * 34 + k]),
                      &Bg[(size_t)k * ldb + col0 + n]);
    }
    wait_async0();
    __syncthreads();

    int cur = 0;
    for (int k0 = 0; k0 < Ktot; k0 += 32) {
        const int nxt = cur ^ 1;
        const bool has_next = (k0 + 32 < Ktot);

        // issue next-slab loads first: A -> registers (global latency hidden
        // behind the WMMA block), B -> LDS via async DMA (no VGPR path).
        float areg[8];
        if (has_next) {
            #pragma unroll
            for (int j = 0; j < 8; ++j) {
                int idx = tid + j * 256;
                int i = idx >> 5, k = idx & 31;
                areg[j] = fA(row0 + i, k0 + 32 + k);
            }
            #pragma unroll
            for (int j = 0; j < 8; ++j) {
                int idx = tid + j * 256;
                int n = idx & 63, k = idx >> 6;
                async_g2l_b32(lds_off_of(&Bs[nxt][n * 34 + k]),
                              &Bg[(size_t)(k0 + 32 + k) * ldb + col0 + n]);
            }
        }

        // ---- compute on current buffers ----
        #pragma unroll
        for (int kk = 0; kk < 32; kk += 4) {
            v2f a  = *(const v2f*)&As[cur][(wm * 16 + l15) * 34 + kk + koff];
            v2f b0 = *(const v2f*)&Bs[cur][(wn * 32 + l15) * 34 + kk + koff];
            v2f b1 = *(const v2f*)&Bs[cur][(wn * 32 + 16 + l15) * 34 + kk + koff];
            acc0 = __builtin_amdgcn_wmma_f32_16x16x4_f32(false, a, false, b0, (short)0, acc0, false, false);
            acc1 = __builtin_amdgcn_wmma_f32_16x16x4_f32(false, a, false, b1, (short)0, acc1, false, false);
        }

        // sink A registers into next buffer
        if (has_next) {
            #pragma unroll
            for (int j = 0; j < 8; ++j) {
                int idx = tid + j * 256;
                int i = idx >> 5, k = idx & 31;
                As[nxt][i * 34 + k] = areg[j];
            }
        }
        wait_async0();
        __syncthreads();
        cur = nxt;
    }

    const int rbase = row0 + wm * 16 + ((lane < 16) ? 0 : 8);
    const int cbase = col0 + wn * 32 + l15;
    #pragma unroll
    for (int i = 0; i < 8; ++i) {
        epi(rbase + i, cbase,      acc0[i]);
        epi(rbase + i, cbase + 16, acc1[i]);
    }
}

// ctx[b] = probs[b] @ H_i[b] :  (512x512)@(512x768), per batch
__global__ void k_gemm_ctx(const float* __restrict__ probs, const float* __restrict__ Hi,
                           float* __restrict__ ctx)
{
    const int b = blockIdx.z;
    AFetchPlain fa{probs + (size_t)b * S_ * S_, S_};
    EpiStore    ep{ctx + (size_t)b * S_ * H_, H_};
    gemm_tile_64x64(fa, Hi + (size_t)b * S_ * H_, H_, S_, blockIdx.y * 64, blockIdx.x * 64, ep);
}

// mhid = relu([ctx | H_j | ctx*H_j] @ Wv1 + bv1) : (2048 x 2304)@(2304 x 768)
__global__ void k_gemm_mhid(const float* __restrict__ ctx, const float* __restrict__ Hj,
                            const float* __restrict__ Wv1, const float* __restrict__ bv1,
                            float* __restrict__ mhid)
{
    AFetchFused fa{ctx, Hj};
    EpiBiasRelu ep{mhid, bv1};
    gemm_tile_64x64(fa, Wv1, O_, 3 * H_, blockIdx.y * 64, blockIdx.x * 64, ep);
}

// out = alpha * (mhid @ Wv2 + bv2) : (2048 x 768)@(768 x 768)
__global__ void k_gemm_out(const float* __restrict__ mhid, const float* __restrict__ Wv2,
                           const float* __restrict__ bv2, const float* __restrict__ alpha,
                           float* __restrict__ out)
{
    AFetchPlain  fa{mhid, O_};
    EpiBiasAlpha ep{out, bv2, alpha};
    gemm_tile_64x64(fa, Wv2, H_, O_, blockIdx.y * 64, blockIdx.x * 64, ep);
}

// ---------------------------------------------------------------------------
extern "C" void kernel_launch(void* const* d_in, const int* in_sizes, int n_in,
                              void* d_out, int out_size, void* d_ws, size_t ws_size,
                              hipStream_t stream)
{
    const float* Hj   = (const float*)d_in[0];
    const float* Hi   = (const float*)d_in[1];
    const float* Wpj  = (const float*)d_in[2];
    const float* Wpi  = (const float*)d_in[3];
    const float* Ws1  = (const float*)d_in[4];
    const float* bs1  = (const float*)d_in[5];
    const float* ws2  = (const float*)d_in[6];
    const float* bs2  = (const float*)d_in[7];
    const float* Wv1  = (const float*)d_in[8];
    const float* bv1  = (const float*)d_in[9];
    const float* Wv2  = (const float*)d_in[10];
    const float* bv2  = (const float*)d_in[11];
    const float* alpha= (const float*)d_in[12];
    const int*   mask = (const int*)d_in[13];

    float* ws   = (float*)d_ws;
    float* Zj   = ws;                                    // B*S*D
    float* Zi   = Zj   + (size_t)B_ * S_ * D_;
    float* tj   = Zi   + (size_t)B_ * S_ * D_;           // B*S*M
    float* ti   = tj   + (size_t)B_ * S_ * M_;
    float* probs= ti   + (size_t)B_ * S_ * M_;           // B*S*S
    float* ctx  = probs+ (size_t)B_ * S_ * S_;           // B*S*H
    float* mhid = ctx  + (size_t)B_ * S_ * H_;           // B*S*O
    float* out  = (float*)d_out;

    k_proj<<<dim3(B_ * S_, 2), 256, 0, stream>>>(Hj, Hi, Wpj, Wpi, Ws1, Zj, Zi, tj, ti);

    k_pair<<<dim3(B_ * S_), 256, 0, stream>>>(Zj, Zi, tj, ti, Ws1, bs1, ws2, bs2, mask, probs);

    k_gemm_ctx<<<dim3(H_ / 64, S_ / 64, B_), 256, 0, stream>>>(probs, Hi, ctx);

    k_gemm_mhid<<<dim3(O_ / 64, (B_ * S_) / 64), 256, 0, stream>>>(ctx, Hj, Wv1, bv1, mhid);

    k_gemm_out<<<dim3(H_ / 64, (B_ * S_) / 64), 256, 0, stream>>>(mhid, Wv2, bv2, alpha, out);
}